// Wiener_KPN_SA_50216757625483
// MI455X (gfx1250) — compile-verified
//
#include <hip/hip_runtime.h>
#include <hip/hip_bf16.h>
#include <math.h>

// ---------------------------------------------------------------------------
// Wiener-KPN CG solve for MI455X (gfx1250).
//   A(d) = conv9x9(d, autocorr(k))  +  exp(reg_w) * L^T L d
// The 9x9 stencil runs on the matrix cores as banded-Toeplitz GEMMs
// (54x V_WMMA_F32_16X16X4_F32 per 16x16 tile). The banded A-blocks are
// batch-constant, so they are prebuilt in WMMA VGPR order and streamed
// branch-free from LDS (no predicated loads / exec gymnastics in the
// WMMA loop). Working set (~70MB) is L2-resident (192MB).
// ---------------------------------------------------------------------------

#define BATCH   8
#define IMG_H   384
#define IMG_W   384
#define HW      (IMG_H * IMG_W)
#define CG_TOL  1e-6f
#define L1_EPS  1e-12f
#define MAXITER 50
#define NBLK    54                      // 9 dx-offsets * 6 K-chunks

typedef __attribute__((ext_vector_type(2))) float v2f;
typedef __attribute__((ext_vector_type(8))) float v8f;

// workspace layout (float offsets)
#define KERN_OFF  0u                          // 8*9*384*384
#define R_OFF     (KERN_OFF + BATCH*9*HW)     // 8*384*384
#define DV_OFF    (R_OFF    + BATCH*HW)
#define Q_OFF     (DV_OFF   + BATCH*HW)
#define KTK_OFF   (Q_OFF    + BATCH*HW)       // 8*81
#define WA_OFF    (KTK_OFF  + BATCH*81)       // 8*54*64 prebuilt WMMA A-blocks
#define SC_OFF    (WA_OFF   + BATCH*NBLK*64)  // scalar block (56 floats)
// scalar block slots
#define S_DELTA 0
#define S_THR   8
#define S_CONV  16
#define S_DTQ   24
#define S_RR    32
#define S_ALPHA 40
#define S_BETA  48

__device__ __forceinline__ int mirror_idx(int i, int n) {
  if (i < 0)  i = -i - 1;          // jnp.pad mode='symmetric'
  if (i >= n) i = 2 * n - 1 - i;
  return i;
}

// --------------------------- one-time setup -------------------------------

// kern = filters_raw / max(sum_t |filters_raw_t|, eps)  (9 taps per pixel)
__global__ void wkpn_normalize_filters(const float* __restrict__ f,
                                       float* __restrict__ kern) {
  int idx = blockIdx.x * 256 + threadIdx.x;       // over BATCH*HW
  int b = idx / HW, p = idx - b * HW;
  size_t base = (size_t)b * 9 * HW + p;
  float s = 0.f;
#pragma unroll
  for (int t = 0; t < 9; ++t) s += fabsf(f[base + (size_t)t * HW]);
  float inv = 1.f / fmaxf(s, L1_EPS);
#pragma unroll
  for (int t = 0; t < 9; ++t) kern[base + (size_t)t * HW] = f[base + (size_t)t * HW] * inv;
}

// 9x9 composite kernel: ktk[d] = sum_v k[v] * k[v+d]  (PSF autocorrelation)
__global__ void wkpn_compute_ktk(const float* __restrict__ k,
                                 float* __restrict__ ktk) {
  int b = blockIdx.x, t = threadIdx.x;
  if (t >= 81) return;
  int dy = t / 9 - 4, dx = t % 9 - 4;
  float s = 0.f;
  for (int u = 0; u < 5; ++u)
    for (int v = 0; v < 5; ++v) {
      int uy = u + dy, vx = v + dx;
      if (uy >= 0 && uy < 5 && vx >= 0 && vx < 5)
        s += k[b * 25 + u * 5 + v] * k[b * 25 + uy * 5 + vx];
    }
  ktk[b * 81 + t] = s;
}

// Prebuild the 54 banded-Toeplitz 16x4 A-blocks in exact WMMA VGPR order:
// wA[b][blk][lane*2 + comp], blk = dx*6 + ch.
//   lanes 0-15 carry K = 4ch+{0,1}, lanes 16-31 carry K = 4ch+{2,3};
//   W[M, t] = ktk[(t-M), dx] when 0 <= t-M <= 8 else 0.
__global__ void wkpn_build_wblocks(const float* __restrict__ ktk,
                                   float* __restrict__ wA) {
  int b = blockIdx.x;
  for (int i = threadIdx.x; i < NBLK * 64; i += 128) {
    int comp = i & 1;
    int lane = (i >> 1) & 31;
    int blk  = i >> 6;
    int dx   = blk / 6, ch = blk - dx * 6;
    int M    = lane & 15;
    int hi   = (lane >> 4) << 1;
    int t    = 4 * ch + hi + comp;
    int dy   = t - M;
    wA[(size_t)b * (NBLK * 64) + i] =
        (dy >= 0 && dy <= 8) ? ktk[b * 81 + dy * 9 + dx] : 0.f;
  }
}

__global__ void wkpn_zero_scalars(float* __restrict__ S) {
  if (threadIdx.x < 56) S[threadIdx.x] = 0.f;
}

// b = K^T y (5x5 correlation, symmetric-mirror boundary); r0 = d0 = b; x0 = 0;
// accumulate delta0 = sum r0^2 per batch.
__global__ void wkpn_b_init(const float* __restrict__ y, const float* __restrict__ k,
                            float* __restrict__ r, float* __restrict__ dvec,
                            float* __restrict__ x, float* __restrict__ S) {
  __shared__ float red;
  int b = blockIdx.z;
  int p = blockIdx.x * 256 + threadIdx.x;
  int pr = p / IMG_W, pc = p - pr * IMG_W;
  const float* yb = y + (size_t)b * HW;
  float s = 0.f;
#pragma unroll
  for (int u = 0; u < 5; ++u)
#pragma unroll
    for (int v = 0; v < 5; ++v)
      s += k[b * 25 + u * 5 + v] *
           yb[mirror_idx(pr + u - 2, IMG_H) * IMG_W + mirror_idx(pc + v - 2, IMG_W)];
  size_t gi = (size_t)b * HW + p;
  r[gi] = s; dvec[gi] = s; x[gi] = 0.f;
  if (threadIdx.x == 0) red = 0.f;
  __syncthreads();
  atomicAdd(&red, s * s);
  __syncthreads();
  if (threadIdx.x == 0) atomicAdd(&S[S_DELTA + b], red);
}

__global__ void wkpn_init_scalars(float* __restrict__ S) {
  int b = threadIdx.x;
  if (b < BATCH) {
    float d = S[S_DELTA + b];
    float thr = CG_TOL * d;
    S[S_THR + b]  = thr;
    S[S_CONV + b] = (d <= thr) ? 1.f : 0.f;
  }
}

// ----------------------------- CG iteration -------------------------------

__global__ void wkpn_pre_iter(float* __restrict__ S) {
  if (threadIdx.x < BATCH) S[S_DTQ + threadIdx.x] = 0.f;
}

// Fused q = A(d) with d.q reduction. Block = 128 threads (4 waves), each wave
// owns one 16x16 output tile; block covers a 32x32 region of one batch image.
__global__ void __launch_bounds__(128)
wkpn_apply_A_dot(const float* __restrict__ dvec, const float* __restrict__ kern,
                 const float* __restrict__ wA, const float* __restrict__ rw,
                 float* __restrict__ q, float* __restrict__ S) {
  __shared__ float  sd[40 * 40];            // d tile + 4-halo (mirror boundary)
  __shared__ float  ss[34 * 34];            // s = L d tile + 1-halo (zero pad)
  __shared__ float2 saA[NBLK * 32];         // prebuilt WMMA A-blocks (13.8KB)
  __shared__ float  sdot;

  const int b    = blockIdx.z;
  const int row0 = blockIdx.y * 32;
  const int col0 = blockIdx.x * 32;
  const int tid  = threadIdx.x;
  const float lam = expf(rw[0]);

  const float* dimg = dvec + (size_t)b * HW;
  const float* kimg = kern + (size_t)b * 9 * HW;
  __builtin_prefetch(kimg + (size_t)row0 * IMG_W + col0, 0, 1);  // global_prefetch_b8

  // stage prebuilt A-blocks (coalesced b64 loads, shared by all 4 waves)
  const float2* wAb = (const float2*)(wA + (size_t)b * (NBLK * 64));
  for (int i = tid; i < NBLK * 32; i += 128) saA[i] = wAb[i];
  // stage d tile, mirrored halo
  for (int i = tid; i < 40 * 40; i += 128) {
    int rr = i / 40, cc = i - rr * 40;
    sd[i] = dimg[mirror_idx(row0 - 4 + rr, IMG_H) * IMG_W +
                 mirror_idx(col0 - 4 + cc, IMG_W)];
  }
  if (tid == 0) sdot = 0.f;
  __syncthreads();

  // s = spatial_conv(d) on tile+1 halo (zero padding, like F.unfold)
  for (int i = tid; i < 34 * 34; i += 128) {
    int rr = i / 34, cc = i - rr * 34;
    int gr = row0 - 1 + rr, gc = col0 - 1 + cc;
    float sv = 0.f;
    if (gr >= 0 && gr < IMG_H && gc >= 0 && gc < IMG_W) {
#pragma unroll
      for (int di = 0; di < 3; ++di)
#pragma unroll
        for (int dj = 0; dj < 3; ++dj) {
          int nr = gr + di - 1, nc = gc + dj - 1;
          float dv = 0.f;
          if (nr >= 0 && nr < IMG_H && nc >= 0 && nc < IMG_W)
            dv = sd[(nr - row0 + 4) * 40 + (nc - col0 + 4)];
          sv += kimg[(size_t)(di * 3 + dj) * HW + (size_t)gr * IMG_W + gc] * dv;
        }
    }
    ss[i] = sv;
  }
  __syncthreads();

  // ---- K^T K on the matrix cores: branch-free A/B feeds, 54 WMMAs ----
  const int wave = tid >> 5, lane = tid & 31;
  const int tr = (wave >> 1) << 4, tc = (wave & 1) << 4;
  const int M  = lane & 15;
  const int hi = (lane >> 4) << 1;            // lanes 16-31 carry K+2,K+3
  const int bcol = tc + M;                    // lane's B column base in sd
  v8f acc = {};
#pragma unroll
  for (int dx = 0; dx < 9; ++dx) {
#pragma unroll
    for (int ch = 0; ch < 6; ++ch) {
      const int t0 = 4 * ch + hi;
      float2 af = saA[(dx * 6 + ch) * 32 + lane];     // ds_load_b64
      v2f a;  a.x = af.x;  a.y = af.y;
      v2f bm;                                         // ds_load_2addr_b32
      bm.x = sd[(tr + t0)     * 40 + (bcol + dx)];
      bm.y = sd[(tr + t0 + 1) * 40 + (bcol + dx)];
      acc = __builtin_amdgcn_wmma_f32_16x16x4_f32(
                false, a, false, bm, (short)0, acc, false, false);
    }
  }

  // ---- add lam * L^T L, store q, accumulate d.q ----
  float local = 0.f;
  const int orow0 = row0 + tr + ((lane >> 4) << 3);  // VGPR r -> rows M=r / M=r+8
  const int ocol  = col0 + tc + M;
#pragma unroll
  for (int rr = 0; rr < 8; ++rr) {
    int orow = orow0 + rr;
    float ltl = 0.f;
#pragma unroll
    for (int di = 0; di < 3; ++di)
#pragma unroll
      for (int dj = 0; dj < 3; ++dj) {
        int pr = orow - (di - 1), pc = ocol - (dj - 1);   // adjoint scatter->gather
        if (pr >= 0 && pr < IMG_H && pc >= 0 && pc < IMG_W)
          ltl += kimg[(size_t)(di * 3 + dj) * HW + (size_t)pr * IMG_W + pc] *
                 ss[(pr - row0 + 1) * 34 + (pc - col0 + 1)];
      }
    float qv = acc[rr] + lam * ltl;
    q[((size_t)b * IMG_H + orow) * IMG_W + ocol] = qv;
    local += qv * sd[(orow - row0 + 4) * 40 + (ocol - col0 + 4)];
  }
  atomicAdd(&sdot, local);
  __syncthreads();
  if (tid == 0) atomicAdd(&S[S_DTQ + b], sdot);
}

__global__ void wkpn_alpha(float* __restrict__ S) {
  int b = threadIdx.x;
  if (b < BATCH) {
    float conv = S[S_CONV + b];
    S[S_ALPHA + b] = (conv != 0.f) ? 0.f : S[S_DELTA + b] / S[S_DTQ + b];
    S[S_RR + b] = 0.f;
  }
}

__global__ void wkpn_update_xr(float* __restrict__ x, const float* __restrict__ dvec,
                               const float* __restrict__ qv, float* __restrict__ r,
                               float* __restrict__ S) {
  __shared__ float red;
  int b = blockIdx.z;
  size_t i = (size_t)b * HW + blockIdx.x * 256 + threadIdx.x;
  float al = S[S_ALPHA + b];
  x[i] += al * dvec[i];
  float rn = r[i] - al * qv[i];
  r[i] = rn;
  if (threadIdx.x == 0) red = 0.f;
  __syncthreads();
  atomicAdd(&red, rn * rn);
  __syncthreads();
  if (threadIdx.x == 0) atomicAdd(&S[S_RR + b], red);
}

__global__ void wkpn_beta(float* __restrict__ S) {
  int b = threadIdx.x;
  if (b < BATCH) {
    float rr = S[S_RR + b];
    S[S_BETA + b] = rr / S[S_DELTA + b];   // reference: beta regardless of conv
    S[S_DELTA + b] = rr;
    if (rr <= S[S_THR + b]) S[S_CONV + b] = 1.f;
  }
}

__global__ void wkpn_update_d(float* __restrict__ dvec, const float* __restrict__ r,
                              const float* __restrict__ S) {
  int b = blockIdx.z;
  size_t i = (size_t)b * HW + blockIdx.x * 256 + threadIdx.x;
  dvec[i] = r[i] + S[S_BETA + b] * dvec[i];
}

// ------------------------------- launcher ---------------------------------

extern "C" void kernel_launch(void* const* d_in, const int* in_sizes, int n_in,
                              void* d_out, int out_size, void* d_ws, size_t ws_size,
                              hipStream_t stream) {
  (void)in_sizes; (void)n_in; (void)out_size; (void)ws_size;
  const float* y     = (const float*)d_in[0];   // [8,1,384,384]
  const float* kblur = (const float*)d_in[1];   // [8,5,5]
  const float* filt  = (const float*)d_in[2];   // [8,9,384,384]
  const float* rw    = (const float*)d_in[3];   // [1]
  float* x  = (float*)d_out;                    // CG iterate x lives in d_out
  float* ws = (float*)d_ws;

  float* kern = ws + KERN_OFF;
  float* r    = ws + R_OFF;
  float* dv   = ws + DV_OFF;
  float* q    = ws + Q_OFF;
  float* ktk  = ws + KTK_OFF;
  float* wAg  = ws + WA_OFF;
  float* S    = ws + SC_OFF;

  wkpn_zero_scalars<<<1, 64, 0, stream>>>(S);
  wkpn_normalize_filters<<<(BATCH * HW) / 256, 256, 0, stream>>>(filt, kern);
  wkpn_compute_ktk<<<BATCH, 96, 0, stream>>>(kblur, ktk);
  wkpn_build_wblocks<<<BATCH, 128, 0, stream>>>(ktk, wAg);
  wkpn_b_init<<<dim3(HW / 256, 1, BATCH), 256, 0, stream>>>(y, kblur, r, dv, x, S);
  wkpn_init_scalars<<<1, 32, 0, stream>>>(S);

  for (int it = 0; it < MAXITER; ++it) {
    wkpn_pre_iter<<<1, 32, 0, stream>>>(S);
    wkpn_apply_A_dot<<<dim3(IMG_W / 32, IMG_H / 32, BATCH), 128, 0, stream>>>(
        dv, kern, wAg, rw, q, S);
    wkpn_alpha<<<1, 32, 0, stream>>>(S);
    wkpn_update_xr<<<dim3(HW / 256, 1, BATCH), 256, 0, stream>>>(x, dv, q, r, S);
    wkpn_beta<<<1, 32, 0, stream>>>(S);
    wkpn_update_d<<<dim3(HW / 256, 1, BATCH), 256, 0, stream>>>(dv, r, S);
  }
}